// SoftDTW_45810121179688
// MI455X (gfx1250) — compile-verified
//
#include <hip/hip_runtime.h>
#include <stdint.h>

#define B_    64
#define N_    512
#define M_    512
#define T_    (N_ + M_ - 1)   // 1023
#define INFV  1e30f
#define PREF  4               // async prefetch depth (power of 2)

#ifndef __has_builtin
#define __has_builtin(x) 0
#endif

#if __has_builtin(__builtin_amdgcn_global_load_async_to_lds_b32)
#define HAVE_ASYNC_LDS 1
#else
#define HAVE_ASYNC_LDS 0
#endif

// ---- waits: use inline asm (memory clobber => real compiler barrier) ----
#if HAVE_ASYNC_LDS
#define ASYNC_WAIT(n) asm volatile("s_wait_asynccnt " #n ::: "memory")
#define DS_WAIT()     asm volatile("s_wait_dscnt 0" ::: "memory")
#else
#define ASYNC_WAIT(n)
#define DS_WAIT()
#endif

typedef __attribute__((address_space(1))) int gint_t;
typedef __attribute__((address_space(3))) int lint_t;

// ---- stage one f32: global -> LDS ----
__device__ __forceinline__ void stage_f32(const float* g, float* l) {
#if HAVE_ASYNC_LDS
    // ASYNCcnt-tracked DMA: copies 4 bytes from global address g to LDS address l  (per-lane addresses)
    __builtin_amdgcn_global_load_async_to_lds_b32(
        (gint_t*)(uintptr_t)g,
        (lint_t*)(uintptr_t)l,
        0, 0);
#else
    *l = __ldg(g);   // synchronous fallback (thread reads back only its own slot)
#endif
}

__global__ __launch_bounds__(512, 1)
void softdtw_fwd_kernel(const float* __restrict__ D, float* __restrict__ out) {
    __shared__ float R[3][N_ + 1];     // rotating anti-diagonals of the DP table
    __shared__ float Db[PREF][N_];     // staged D anti-diagonal ring buffer

    const int b   = blockIdx.x;
    const int tid = threadIdx.x;                  // row i = tid + 1
    const float* row = D + (size_t)b * (N_ * M_) + (size_t)tid * M_;

    // ---- init DP diagonals: p2 = {0, INF...}, p1 = {INF...} ----
    for (int k = tid; k <= N_; k += N_) {
        R[0][k] = (k == 0) ? 0.0f : INFV;
        R[1][k] = INFV;
        R[2][k] = INFV;
    }

    // ---- prime the async ring with steps 0..PREF-1 ----
    #pragma unroll
    for (int p = 0; p < PREF; ++p) {
        int jm1 = p - tid;
        jm1 = jm1 < 0 ? 0 : (jm1 > (M_ - 1) ? (M_ - 1) : jm1);   // clamped like jnp.clip
        stage_f32(row + jm1, &Db[p][tid]);
    }
    __syncthreads();

    int t = 0;
    // ================= main pipelined loop =================
    #pragma unroll 1
    for (; t < T_ - PREF; ++t) {
        ASYNC_WAIT(3);                       // in-order returns => load(t) complete
        const float Dval = Db[t & (PREF - 1)][tid];
        DS_WAIT();                           // LDS read done before slot is re-filled
        {   // issue load for step t+PREF into the slot just consumed
            const int tp  = t + PREF;
            int jm1 = tp - tid;
            jm1 = jm1 < 0 ? 0 : (jm1 > (M_ - 1) ? (M_ - 1) : jm1);
            stage_f32(row + jm1, &Db[tp & (PREF - 1)][tid]);
        }
        const float* p2 = R[t % 3];
        const float* p1 = R[(t + 1) % 3];
        float*       nw = R[(t + 2) % 3];

        const float r0 = p2[tid];            // R[i-1][j-1]
        const float r1 = p1[tid];            // R[i-1][j]
        const float r2 = p1[tid + 1];        // R[i][j-1]
        const float m  = fminf(r0, fminf(r1, r2));
        const float s  = __expf(m - r0) + __expf(m - r1) + __expf(m - r2);
        const float sm = m - __logf(s);      // == -log(e^-r0+e^-r1+e^-r2), gamma=1
        const bool  valid = ((unsigned)(t - tid)) < (unsigned)M_;
        nw[tid + 1] = valid ? (Dval + sm) : INFV;
        if (tid == 0) nw[0] = INFV;
        __syncthreads();
    }
    // ================= drain tail (no more issues) =================
    #pragma unroll 1
    for (; t < T_; ++t) {
        ASYNC_WAIT(0);
        const float Dval = Db[t & (PREF - 1)][tid];
        const float* p2 = R[t % 3];
        const float* p1 = R[(t + 1) % 3];
        float*       nw = R[(t + 2) % 3];
        const float r0 = p2[tid];
        const float r1 = p1[tid];
        const float r2 = p1[tid + 1];
        const float m  = fminf(r0, fminf(r1, r2));
        const float s  = __expf(m - r0) + __expf(m - r1) + __expf(m - r2);
        const float sm = m - __logf(s);
        const bool  valid = ((unsigned)(t - tid)) < (unsigned)M_;
        nw[tid + 1] = valid ? (Dval + sm) : INFV;
        if (tid == 0) nw[0] = INFV;
        __syncthreads();
    }

    // final "new" diagonal is R[(T_-1+2)%3]; answer is its entry N
    if (tid == 0) out[b] = R[(T_ + 1) % 3][N_];
}

extern "C" void kernel_launch(void* const* d_in, const int* in_sizes, int n_in,
                              void* d_out, int out_size, void* d_ws, size_t ws_size,
                              hipStream_t stream) {
    (void)in_sizes; (void)n_in; (void)out_size; (void)d_ws; (void)ws_size;
    const float* D = (const float*)d_in[0];
    float* out     = (float*)d_out;
    softdtw_fwd_kernel<<<B_, N_, 0, stream>>>(D, out);
}